// TalkerDecodeForExport_66864050864551
// MI455X (gfx1250) — compile-verified
//
#include <hip/hip_runtime.h>
#include <hip/hip_bf16.h>
#include <math.h>

// ---- model dims (match reference) ----
#define NLAYERS 28
#define HDIM    1024
#define NHEADS  16
#define NKVH    8
#define DHEAD   128
#define FFDIM   3072
#define VOCAB   3072
#define SEQL    2048

#define KC      8      // K-split chunks per GEMV (occupancy for HBM streaming)
#define ASPLIT  8      // T-split chunks for flash-decode attention

typedef float v2f __attribute__((ext_vector_type(2)));
typedef float v8f __attribute__((ext_vector_type(8)));

// ---------------------------------------------------------------------------
// RMSNorm: out = in * rsqrt(mean(in^2)+eps) * w    (single block)
// ---------------------------------------------------------------------------
__global__ void rms_kernel(const float* __restrict__ in, const float* __restrict__ w,
                           float* __restrict__ out, int n) {
    __shared__ float sbuf[256];
    float acc = 0.f;
    for (int i = threadIdx.x; i < n; i += blockDim.x) { float t = in[i]; acc += t * t; }
    sbuf[threadIdx.x] = acc; __syncthreads();
    for (int s = blockDim.x >> 1; s > 0; s >>= 1) {
        if (threadIdx.x < (unsigned)s) sbuf[threadIdx.x] += sbuf[threadIdx.x + s];
        __syncthreads();
    }
    float r = rsqrtf(sbuf[0] / (float)n + 1e-6f);
    for (int i = threadIdx.x; i < n; i += blockDim.x) out[i] = in[i] * r * w[i];
}

// ---------------------------------------------------------------------------
// K-split GEMV partial via V_WMMA_F32_16X16X4_F32.
//   ypart[ck*N + col] = x[k0:k0+klen] @ W[k0:k0+klen, col]   (ck = blockIdx.y)
// One wave per 16-column tile, 4 waves (64 cols) per block, KC chunks in y.
// A: row0 = x[k..k+3] (lanes 0 & 16), rows 1..15 zero.  B: 4x16 weight tile,
// lane n holds {W[r][n], W[r+1][n]} with r = k + 2*(lane>=16) -> coalesced
// 64B row segments.  Result for column n lives in C[0] on lanes 0..15.
// Unroll x8 keeps 16 weight loads (+ prefetches) in flight per wait window so
// each wave can stream HBM at full rate instead of stalling per 256B.
// ---------------------------------------------------------------------------
__global__ void gemv_wmma_part(const float* __restrict__ x, const float* __restrict__ W,
                               float* __restrict__ ypart, int K, int N) {
    extern __shared__ float xs[];                 // klen floats
    const int klen = K / KC;
    const int k0   = blockIdx.y * klen;
    for (int i = threadIdx.x; i < klen; i += blockDim.x) xs[i] = x[k0 + i];
    __syncthreads();

    const int lane = threadIdx.x & 31;
    const int wave = threadIdx.x >> 5;
    const int tile = blockIdx.x * 4 + wave;       // 16-column tile index
    const int col  = tile * 16 + (lane & 15);
    const int hi   = lane >> 4;                   // 0: lanes 0-15, 1: lanes 16-31
    const float* Wc = W + (size_t)k0 * (size_t)N;

    v8f c = {};
    #pragma unroll 8
    for (int k = 0; k < klen; k += 4) {
        if (k + 16 < klen) {                      // run a fixed distance ahead
            __builtin_prefetch(&Wc[(size_t)(k + 16 + hi * 2) * (size_t)N + col], 0, 1);
        }
        v2f a;
        a.x = (lane == 0) ? xs[k + 0] : ((lane == 16) ? xs[k + 2] : 0.f);
        a.y = (lane == 0) ? xs[k + 1] : ((lane == 16) ? xs[k + 3] : 0.f);
        const size_t r0 = (size_t)(k + hi * 2) * (size_t)N + (size_t)col;
        v2f b;
        b.x = Wc[r0];
        b.y = Wc[r0 + (size_t)N];
        c = __builtin_amdgcn_wmma_f32_16x16x4_f32(false, a, false, b, (short)0, c,
                                                  false, false);
    }
    if (lane < 16) ypart[(size_t)blockIdx.y * (size_t)N + col] = c[0];
}

// ---------------------------------------------------------------------------
// y[i] = sum_c part[c][i] (+ resid[i])     fixed-order -> deterministic
// ---------------------------------------------------------------------------
__global__ void combine_add_kernel(const float* __restrict__ part, int N,
                                   const float* __restrict__ resid, float* __restrict__ y) {
    int i = blockIdx.x * blockDim.x + threadIdx.x;
    if (i < N) {
        float s = 0.f;
        #pragma unroll
        for (int c = 0; c < KC; ++c) s += part[(size_t)c * N + i];
        if (resid) s += resid[i];
        y[i] = s;
    }
}

// ---------------------------------------------------------------------------
// act[i] = silu(sum_c g[c][i]) * sum_c u[c][i]
// ---------------------------------------------------------------------------
__global__ void combine_silu_kernel(const float* __restrict__ gp, const float* __restrict__ up,
                                    int N, float* __restrict__ out) {
    int i = blockIdx.x * blockDim.x + threadIdx.x;
    if (i < N) {
        float g = 0.f, u = 0.f;
        #pragma unroll
        for (int c = 0; c < KC; ++c) {
            g += gp[(size_t)c * N + i];
            u += up[(size_t)c * N + i];
        }
        out[i] = (g / (1.f + __expf(-g))) * u;
    }
}

// ---------------------------------------------------------------------------
// Per-head RMSNorm (q/k) + mRoPE + scatter new k/v token into present cache.
// Blocks 0..15: q heads (in-place in qbuf). Blocks 16..23: k heads -> cache.
// ---------------------------------------------------------------------------
__global__ void rope_scatter_kernel(float* __restrict__ qbuf, const float* __restrict__ kbuf,
                                    const float* __restrict__ vbuf,
                                    const float* __restrict__ qnw, const float* __restrict__ knw,
                                    const int* __restrict__ pos_ids, const int* __restrict__ cache_pos,
                                    float* __restrict__ keysL, float* __restrict__ valsL) {
    __shared__ float vs[DHEAD];
    __shared__ float sbuf[DHEAD];
    const int b = blockIdx.x, j = threadIdx.x;

    const float* src; const float* nw;
    if (b < NHEADS) { src = qbuf + b * DHEAD;            nw = qnw; }
    else            { src = kbuf + (b - NHEADS) * DHEAD; nw = knw; }

    float val = src[j];
    sbuf[j] = val * val; __syncthreads();
    for (int s = DHEAD >> 1; s > 0; s >>= 1) {
        if (j < s) sbuf[j] += sbuf[j + s];
        __syncthreads();
    }
    float r  = rsqrtf(sbuf[0] * (1.f / DHEAD) + 1e-6f);
    float xn = val * r * nw[j];
    vs[j] = xn; __syncthreads();
    float rot = (j < 64) ? -vs[j + 64] : vs[j - 64];

    // mrope sections (doubled): [0,32)->t plane, [32,80)->h, [80,128)->w
    int plane = (j < 32) ? 0 : ((j < 80) ? 1 : 2);
    float p    = (float)pos_ids[plane];
    float invf = powf(1.0e6f, -(float)(j & 63) * (1.0f / 64.0f)); // THETA^-(i/64)
    float ang  = p * invf;
    float outv = xn * cosf(ang) + rot * sinf(ang);

    if (b < NHEADS) {
        qbuf[b * DHEAD + j] = outv;
    } else {
        const int hk  = b - NHEADS;
        const int pos = cache_pos[0];
        keysL[((size_t)hk * SEQL + pos) * DHEAD + j] = outv;
        valsL[((size_t)hk * SEQL + pos) * DHEAD + j] = vbuf[hk * DHEAD + j];
    }
}

// ---------------------------------------------------------------------------
// Flash-decode partial: block = (q head, T-chunk), 128 threads.
// Emits per-chunk (m, s, unnormalized o) for deterministic merge.
// Causal mask handled by limiting T (mask is 0 for t<=pos, -1e9 otherwise).
// ---------------------------------------------------------------------------
__global__ void attn_part_kernel(const float* __restrict__ qbuf, const float* __restrict__ keysL,
                                 const float* __restrict__ valsL,
                                 float* __restrict__ m_part, float* __restrict__ s_part,
                                 float* __restrict__ o_part, const int* __restrict__ cache_pos) {
    __shared__ __align__(16) float qs[DHEAD];
    __shared__ float sc[(SEQL + ASPLIT - 1) / ASPLIT];   // 256 max
    __shared__ float red[DHEAD];
    const int hq = blockIdx.x, cs = blockIdx.y, tid = threadIdx.x;
    const int kv = hq >> 1;                       // G = NH/NKV = 2
    const int T  = cache_pos[0] + 1;
    const int clen = (T + ASPLIT - 1) / ASPLIT;
    const int t0 = cs * clen;
    const int t1 = (t0 + clen < T) ? (t0 + clen) : T;
    const float scale = 0.08838834764831845f;     // 1/sqrt(128)

    qs[tid] = qbuf[hq * DHEAD + tid];
    __syncthreads();

    const float* kb = keysL + (size_t)kv * SEQL * DHEAD;
    const float* vb = valsL + (size_t)kv * SEQL * DHEAD;
    const float4* qs4 = (const float4*)qs;

    float lmax = -1e30f;
    for (int t = t0 + tid; t < t1; t += DHEAD) {
        const float4* kr = (const float4*)(kb + (size_t)t * DHEAD);
        float d = 0.f;
        #pragma unroll
        for (int i = 0; i < DHEAD / 4; ++i) {
            float4 a = qs4[i], bb = kr[i];
            d += a.x * bb.x + a.y * bb.y + a.z * bb.z + a.w * bb.w;
        }
        d *= scale;
        sc[t - t0] = d;
        lmax = fmaxf(lmax, d);
    }
    red[tid] = lmax; __syncthreads();
    for (int s = DHEAD >> 1; s > 0; s >>= 1) {
        if (tid < s) red[tid] = fmaxf(red[tid], red[tid + s]);
        __syncthreads();
    }
    float m = red[0]; __syncthreads();

    float lsum = 0.f;
    for (int t = t0 + tid; t < t1; t += DHEAD) {
        float p = __expf(sc[t - t0] - m);
        sc[t - t0] = p;
        lsum += p;
    }
    red[tid] = lsum; __syncthreads();
    for (int s = DHEAD >> 1; s > 0; s >>= 1) {
        if (tid < s) red[tid] += red[tid + s];
        __syncthreads();
    }
    float ssum = red[0];
    __syncthreads();

    float acc = 0.f;                              // thread tid owns output dim d=tid
    for (int t = t0; t < t1; ++t) acc += sc[t - t0] * vb[(size_t)t * DHEAD + tid];

    const int pi = hq * ASPLIT + cs;
    if (tid == 0) { m_part[pi] = m; s_part[pi] = ssum; }
    o_part[(size_t)pi * DHEAD + tid] = acc;       // unnormalized
}

// ---------------------------------------------------------------------------
// Merge flash-decode partials in fixed chunk order (deterministic).
// ---------------------------------------------------------------------------
__global__ void attn_combine_kernel(const float* __restrict__ m_part, const float* __restrict__ s_part,
                                    const float* __restrict__ o_part, float* __restrict__ obuf) {
    const int hq = blockIdx.x, tid = threadIdx.x;   // 128 threads
    float M = -1e30f;
    #pragma unroll
    for (int c = 0; c < ASPLIT; ++c) M = fmaxf(M, m_part[hq * ASPLIT + c]);
    float S = 0.f, acc = 0.f;
    #pragma unroll
    for (int c = 0; c < ASPLIT; ++c) {
        float w = __expf(m_part[hq * ASPLIT + c] - M);
        S   += w * s_part[hq * ASPLIT + c];
        acc += w * o_part[(size_t)(hq * ASPLIT + c) * DHEAD + tid];
    }
    obuf[hq * DHEAD + tid] = acc / S;
}

// ---------------------------------------------------------------------------
extern "C" void kernel_launch(void* const* d_in, const int* in_sizes, int n_in,
                              void* d_out, int out_size, void* d_ws, size_t ws_size,
                              hipStream_t stream) {
    (void)in_sizes; (void)n_in; (void)out_size; (void)ws_size;

    const float* inputs_embeds = (const float*)d_in[0];
    const int*   position_ids  = (const int*)d_in[1];
    const int*   cache_pos     = (const int*)d_in[2];
    // d_in[3] attention_mask: implied by cache_pos (0 for t<=pos, -1e9 after)
    const float* past_keys     = (const float*)d_in[4];
    const float* past_values   = (const float*)d_in[5];
    const float* Wq    = (const float*)d_in[6];
    const float* Wk    = (const float*)d_in[7];
    const float* Wv    = (const float*)d_in[8];
    const float* Wo    = (const float*)d_in[9];
    const float* qnw   = (const float*)d_in[10];
    const float* knw   = (const float*)d_in[11];
    const float* ln1   = (const float*)d_in[12];
    const float* ln2   = (const float*)d_in[13];
    const float* Wg    = (const float*)d_in[14];
    const float* Wu    = (const float*)d_in[15];
    const float* Wd    = (const float*)d_in[16];
    const float* normw = (const float*)d_in[17];
    const float* Whead = (const float*)d_in[18];

    // workspace layout (floats)
    float* ws    = (float*)d_ws;
    float* h     = ws;                        // 1024
    float* x     = h    + HDIM;               // 1024
    float* q     = x    + HDIM;               // 2048
    float* k     = q    + NHEADS * DHEAD;     // 1024
    float* v     = k    + NKVH * DHEAD;       // 1024
    float* o     = v    + NKVH * DHEAD;       // 2048
    float* x2    = o    + NHEADS * DHEAD;     // 1024
    float* act   = x2   + HDIM;               // 3072
    float* part0 = act  + FFDIM;              // KC*3072
    float* gpart = part0 + KC * FFDIM;        // KC*3072
    float* upart = gpart + KC * FFDIM;        // KC*3072
    float* mpart = upart + KC * FFDIM;        // 128
    float* spart = mpart + NHEADS * ASPLIT;   // 128
    float* opart = spart + NHEADS * ASPLIT;   // 16384

    // output layout: [logits(3072) | h(1024) | present_keys | present_values]
    float* out      = (float*)d_out;
    float* logits   = out;
    float* hout     = out + VOCAB;
    float* keys_out = out + VOCAB + HDIM;
    const size_t KVSZ = (size_t)NLAYERS * NKVH * SEQL * DHEAD;
    float* vals_out = keys_out + KVSZ;

    // bulk present-KV copy (dominant output traffic) + residual stream init
    hipMemcpyAsync(h, inputs_embeds, HDIM * sizeof(float), hipMemcpyDeviceToDevice, stream);
    hipMemcpyAsync(keys_out, past_keys,   KVSZ * sizeof(float), hipMemcpyDeviceToDevice, stream);
    hipMemcpyAsync(vals_out, past_values, KVSZ * sizeof(float), hipMemcpyDeviceToDevice, stream);

    for (int l = 0; l < NLAYERS; ++l) {
        const float* Wql = Wq + (size_t)l * HDIM * (NHEADS * DHEAD);
        const float* Wkl = Wk + (size_t)l * HDIM * (NKVH * DHEAD);
        const float* Wvl = Wv + (size_t)l * HDIM * (NKVH * DHEAD);
        const float* Wol = Wo + (size_t)l * (NHEADS * DHEAD) * HDIM;
        const float* Wgl = Wg + (size_t)l * HDIM * FFDIM;
        const float* Wul = Wu + (size_t)l * HDIM * FFDIM;
        const float* Wdl = Wd + (size_t)l * FFDIM * HDIM;
        float* keysL = keys_out + (size_t)l * NKVH * SEQL * DHEAD;
        float* valsL = vals_out + (size_t)l * NKVH * SEQL * DHEAD;

        rms_kernel<<<1, 256, 0, stream>>>(h, ln1 + l * HDIM, x, HDIM);

        // QKV projections (K-split WMMA GEMV + deterministic combine)
        gemv_wmma_part<<<dim3((NHEADS * DHEAD) / 64, KC), 128, (HDIM / KC) * 4, stream>>>(x, Wql, part0, HDIM, NHEADS * DHEAD);
        combine_add_kernel<<<(NHEADS * DHEAD) / 256, 256, 0, stream>>>(part0, NHEADS * DHEAD, nullptr, q);
        gemv_wmma_part<<<dim3((NKVH * DHEAD) / 64, KC), 128, (HDIM / KC) * 4, stream>>>(x, Wkl, part0, HDIM, NKVH * DHEAD);
        combine_add_kernel<<<(NKVH * DHEAD) / 256, 256, 0, stream>>>(part0, NKVH * DHEAD, nullptr, k);
        gemv_wmma_part<<<dim3((NKVH * DHEAD) / 64, KC), 128, (HDIM / KC) * 4, stream>>>(x, Wvl, part0, HDIM, NKVH * DHEAD);
        combine_add_kernel<<<(NKVH * DHEAD) / 256, 256, 0, stream>>>(part0, NKVH * DHEAD, nullptr, v);

        rope_scatter_kernel<<<NHEADS + NKVH, DHEAD, 0, stream>>>(q, k, v, qnw + l * DHEAD, knw + l * DHEAD,
                                                                 position_ids, cache_pos, keysL, valsL);

        // flash-decode attention: T-split partials + deterministic merge
        attn_part_kernel<<<dim3(NHEADS, ASPLIT), DHEAD, 0, stream>>>(q, keysL, valsL, mpart, spart, opart, cache_pos);
        attn_combine_kernel<<<NHEADS, DHEAD, 0, stream>>>(mpart, spart, opart, o);

        // O projection + residual
        gemv_wmma_part<<<dim3(HDIM / 64, KC), 128, ((NHEADS * DHEAD) / KC) * 4, stream>>>(o, Wol, part0, NHEADS * DHEAD, HDIM);
        combine_add_kernel<<<HDIM / 256, 256, 0, stream>>>(part0, HDIM, h, h);

        // MLP
        rms_kernel<<<1, 256, 0, stream>>>(h, ln2 + l * HDIM, x2, HDIM);
        gemv_wmma_part<<<dim3(FFDIM / 64, KC), 128, (HDIM / KC) * 4, stream>>>(x2, Wgl, gpart, HDIM, FFDIM);
        gemv_wmma_part<<<dim3(FFDIM / 64, KC), 128, (HDIM / KC) * 4, stream>>>(x2, Wul, upart, HDIM, FFDIM);
        combine_silu_kernel<<<FFDIM / 256, 256, 0, stream>>>(gpart, upart, FFDIM, act);
        gemv_wmma_part<<<dim3(HDIM / 64, KC), 128, (FFDIM / KC) * 4, stream>>>(act, Wdl, part0, FFDIM, HDIM);
        combine_add_kernel<<<HDIM / 256, 256, 0, stream>>>(part0, HDIM, h, h);
    }

    rms_kernel<<<1, 256, 0, stream>>>(h, normw, hout, HDIM);
    gemv_wmma_part<<<dim3(VOCAB / 64, KC), 128, (HDIM / KC) * 4, stream>>>(hout, Whead, part0, HDIM, VOCAB);
    combine_add_kernel<<<VOCAB / 256, 256, 0, stream>>>(part0, VOCAB, nullptr, logits);
}